// Norm1dBatched_31207232373392
// MI455X (gfx1250) — compile-verified
//
#include <hip/hip_runtime.h>
#include <stdint.h>

#define ALPHA    0.999f
#define ALPHA_B  0.59914228f        /* 0.999^512 (double-precision derived) */
#define LOG2_A  (-0.001443293056f)  /* log2(0.999) */
#define EPSV     1e-5f
#define BB       512
#define LL       8192
#define TR       16                 /* rows per LDS tile */
#define NT       (BB / TR)          /* 32 tiles */
#define TC       256                /* columns per block */
#define NBUF     3                  /* async pipeline depth (3*80KB = 240KB LDS) */

typedef __attribute__((ext_vector_type(2))) float v2f;
typedef __attribute__((ext_vector_type(8))) float v8f;

// ---------------- CDNA5 async-copy helpers (gfx1250) ------------------------
__device__ __forceinline__ void async_ld_b128(uint32_t lds_byte, const float* g) {
  // per-lane: copies 16 bytes from global address g to LDS byte address lds_byte; tracked on ASYNCcnt
  asm volatile("global_load_async_to_lds_b128 %0, %1, off"
               :: "v"(lds_byte), "v"((uint64_t)(uintptr_t)g)
               : "memory");
}

#if __has_builtin(__builtin_amdgcn_s_wait_asynccnt)
#define WAIT_ASYNC(n) __builtin_amdgcn_s_wait_asynccnt(n)
#else
#define WAIT_ASYNC(n) asm volatile("s_wait_asynccnt " #n ::: "memory")
#endif

// ---------------------------------------------------------------------------
// Prologue: per column l,
//   ws_m[l] = sum_{j=1..511} alpha^(512-j) * m_p[j,l]
//   ws_v[l] = sum_{j=1..511} alpha^(512-j) * var_p[j,l]
// via V_WMMA_F32_16X16X4_F32: one wave handles 16 columns. A = 16 identical
// rows of geometric weights, updated incrementally (w *= alpha^-4 per K-chunk,
// no transcendentals in the loop). Row j=0 is included with weight alpha^512
// and subtracted exactly afterwards, eliminating the j==0 branch.
// ---------------------------------------------------------------------------
__global__ void __launch_bounds__(32)
prologue_wmma(const float* __restrict__ m_p, const float* __restrict__ var_p,
              float* __restrict__ ws_m, float* __restrict__ ws_v) {
  const int lane = threadIdx.x;              // 0..31
  const int hi   = lane >> 4;                // 0: lanes 0-15, 1: lanes 16-31
  const int coln = blockIdx.x * 16 + (lane & 15);

  v8f accM = {};
  v8f accV = {};

  // A-matrix 16x4 fp32 layout: v0 = A[m, 2*hi], v1 = A[m, 2*hi+1].
  // At K-chunk r this lane needs w(j) = alpha^(512-j) for j = r+2*hi, r+1+2*hi.
  v2f a;
  a.x = __builtin_exp2f((float)(BB     - 2 * hi) * LOG2_A);
  a.y = __builtin_exp2f((float)(BB - 1 - 2 * hi) * LOG2_A);
  const float c4 = __builtin_exp2f(-4.0f * LOG2_A);   // alpha^-4

  for (int r = 0; r < BB; r += 4) {
    const int j0 = r + 2 * hi;
    const int j1 = j0 + 1;

    v2f bM, bV;
    // B-matrix 4x16 fp32: v0 = B[2*hi, n], v1 = B[2*hi+1, n], n = lane%16
    bM.x = m_p[(size_t)j0 * LL + coln];
    bM.y = m_p[(size_t)j1 * LL + coln];
    bV.x = var_p[(size_t)j0 * LL + coln];
    bV.y = var_p[(size_t)j1 * LL + coln];

    accM = __builtin_amdgcn_wmma_f32_16x16x4_f32(false, a, false, bM,
                                                 (short)0, accM, false, false);
    accV = __builtin_amdgcn_wmma_f32_16x16x4_f32(false, a, false, bV,
                                                 (short)0, accV, false, false);
    a = a * c4;                               // advance weights by 4 K-steps
  }

  // All 16 rows of D are identical (A rows identical); row 0 lives in
  // acc[0], lanes 0..15 (N = lane). Remove the spurious j=0 contribution.
  if (lane < 16) {
    ws_m[coln] = accM[0] - ALPHA_B * m_p[coln];
    ws_v[coln] = accV[0] - ALPHA_B * var_p[coln];
  }
}

// ---------------------------------------------------------------------------
// Main fused scan: one thread per feature column; 512-step linear recurrence
// for both the mean and variance EMAs, fed by a triple-buffered
// global->LDS async pipeline (5 arrays x 16 rows x 256 cols per tile).
// ---------------------------------------------------------------------------
__global__ void __launch_bounds__(TC)
scan_kernel(const float* __restrict__ x,    const float* __restrict__ m_p,
            const float* __restrict__ var_p,const float* __restrict__ m,
            const float* __restrict__ var,  const float* __restrict__ ws_m,
            const float* __restrict__ ws_v, float* __restrict__ out) {
  extern __shared__ float smem[];            // [NBUF][5][TR][TC]
  const int tid      = threadIdx.x;
  const int colBase  = blockIdx.x * TC;
  const int col      = colBase + tid;

  const float* arrs[5] = {x, m_p, var_p, m, var};

  const uint32_t smemB    = (uint32_t)(uint64_t)(uintptr_t)smem;  // LDS byte addr
  const uint32_t bufBytes = 5u * TR * TC * 4u;

  const int rsub = tid >> 6;                 // 0..3  (row slot)
  const int c4   = (tid & 63) << 2;          // 0..252 (4-col slot)

  // Issue one tile (16 rows x 256 cols x 5 arrays) = 20 b128 asyncs/thread.
  auto issue = [&](int b, int r0) {
    const uint32_t lb = smemB + (uint32_t)b * bufBytes;
#pragma unroll
    for (int a = 0; a < 5; ++a) {
      const float* g  = arrs[a] + (size_t)(r0 + rsub) * LL + colBase + c4;
      const uint32_t l = lb + (uint32_t)(((a * TR + rsub) * TC + c4) * 4);
#pragma unroll
      for (int rr = 0; rr < 4; ++rr) {
        async_ld_b128(l + (uint32_t)(rr * 4 * TC * 4),
                      g + (size_t)(rr * 4) * LL);
      }
    }
  };

  // stale state for t == 0 comes from the *last* row of the stream states
  const float mlast = m  [(size_t)(BB - 1) * LL + col];
  const float vlast = var[(size_t)(BB - 1) * LL + col];

  float tm = ws_m[col];                      // becomes tmp_m[0] after +x[0]
  float tv = ws_v[col];                      // becomes tmp_v[0] after +var_cur[0]
  const float oma = 1.0f - ALPHA;

  issue(0, 0 * TR);                          // prime two tiles
  issue(1, 1 * TR);

  float pcm = 0.0f, pcv = 0.0f;              // curr_m[t-1], curr_v[t-1]
  for (int tile = 0; tile < NT; ++tile) {
    if (tile + 2 < NT) {
      issue((tile + 2) % NBUF, (tile + 2) * TR);
      WAIT_ASYNC(40);                        // tile done; 2 tiles still in flight
    } else if (tile + 1 < NT) {
      WAIT_ASYNC(20);                        // drain: 1 tile in flight
    } else {
      WAIT_ASYNC(0);
    }
    __syncthreads();

    const float* buf = smem + (size_t)(tile % NBUF) * (5 * TR * TC);
#pragma unroll
    for (int rr = 0; rr < TR; ++rr) {
      const int t = tile * TR + rr;
      const float xx = buf[(0 * TR + rr) * TC + tid];
      const float mp = buf[(1 * TR + rr) * TC + tid];
      const float vp = buf[(2 * TR + rr) * TC + tid];
      const float mm = buf[(3 * TR + rr) * TC + tid];
      const float vv = buf[(4 * TR + rr) * TC + tid];

      float sm, sv;
      if (t == 0) { tm = tm + xx;                          sm = mlast; }
      else        { tm = ALPHA * tm - ALPHA_B * mp + xx;   sm = pcm;   }
      const float cm   = ALPHA_B * mm + oma * tm;          // curr_m[t]
      const float cent = xx - sm;
      const float vc   = ALPHA * cent * cent;              // var_cur[t]
      if (t == 0) { tv = tv + vc;                          sv = vlast; }
      else        { tv = ALPHA * tv - ALPHA_B * vp + vc;   sv = pcv;   }
      const float cv   = ALPHA_B * vv + oma * tv;          // curr_v[t]

      out[(size_t)t * LL + col] = cent * rsqrtf(sv + EPSV);
      pcm = cm;
      pcv = cv;
    }
    __syncthreads();                         // buffer re-filled next iteration
  }
}

// ---------------------------------------------------------------------------
extern "C" void kernel_launch(void* const* d_in, const int* in_sizes, int n_in,
                              void* d_out, int out_size, void* d_ws, size_t ws_size,
                              hipStream_t stream) {
  (void)in_sizes; (void)n_in; (void)out_size; (void)ws_size;
  const float* x     = (const float*)d_in[0];
  const float* m_p   = (const float*)d_in[1];
  const float* var_p = (const float*)d_in[2];
  const float* m     = (const float*)d_in[3];
  const float* var   = (const float*)d_in[4];
  float* out  = (float*)d_out;
  float* ws_m = (float*)d_ws;          // 8192 floats
  float* ws_v = ws_m + LL;             // 8192 floats (64 KB total scratch)

  prologue_wmma<<<LL / 16, 32, 0, stream>>>(m_p, var_p, ws_m, ws_v);

  const size_t shmem = (size_t)NBUF * 5 * TR * TC * sizeof(float);  // 240 KB
  scan_kernel<<<LL / TC, TC, shmem, stream>>>(x, m_p, var_p, m, var,
                                              ws_m, ws_v, out);
}